// CARAFEPack_68341519614298
// MI455X (gfx1250) — compile-verified
//
#include <hip/hip_runtime.h>
#include <hip/hip_bf16.h>
#include <math.h>

// ---------------------------------------------------------------------------
// CARAFE forward for MI455X (gfx1250, wave32, WMMA).
// Shapes (fixed by the reference): N=4, C=256, H=W=64, CC=64, K=5, SF=2,
// enc_out = 100, Ho=Wo=128.
//
// Pipeline (all through d_ws, L2-resident):
//   [T] transpose_we  : Wet(112,576) tap-major-K copy of We (zero-padded rows)
//   [A] compress_wmma : comp(4,64,64,64)   = Wc(64,256) x X       (f32 WMMA)
//   [B] encoder_wmma  : m(4,100,64,64)     = Wet(112,576) x im2col(comp)
//   [C] softmax_mask  : mask(4,25,128,128) = softmax_k(pixelshuffle(m))
//   [D] reassemble    : out(4,256,128,128) = sum_k mask * x 5x5 patches
// ---------------------------------------------------------------------------

typedef float v2f __attribute__((ext_vector_type(2)));
typedef float v8f __attribute__((ext_vector_type(8)));

#define HW   4096      // 64*64
#define HOWO 16384     // 128*128

// ---------------------------------------------------------------------------
// [T] Repack We (100,64,3,3) -> Wet (112, 576) with K reordered tap-major:
//     Wet[e][t*64 + cc] = We[e][cc*9 + t],  rows 100..111 zeroed.
// Lets the encoder compute tap geometry once per 64 K-steps and keeps the
// A fragment a contiguous b64 load with no ragged-M masking math.
// ---------------------------------------------------------------------------
__global__ __launch_bounds__(256) void transpose_we(
    const float* __restrict__ We, float* __restrict__ Wet)
{
    int idx = blockIdx.x * blockDim.x + threadIdx.x;   // [0, 112*576)
    int e = idx / 576;
    int r = idx - e * 576;
    int t  = r >> 6;          // tap [0,9)
    int cc = r & 63;          // channel [0,64)
    Wet[idx] = (e < 100) ? We[(size_t)e * 576 + cc * 9 + t] : 0.0f;
}

// ---------------------------------------------------------------------------
// [A] 1x1 conv compressor: comp[n,m,p] = sum_k Wc[m,k] * x[n,k,p] + bc[m]
// One wave -> 16(M) x 16(pixel) tile, K=256 in steps of 4.
// ---------------------------------------------------------------------------
__global__ __launch_bounds__(256) void compress_wmma(
    const float* __restrict__ x, const float* __restrict__ Wc,
    const float* __restrict__ bc, float* __restrict__ comp)
{
    const int lane = threadIdx.x & 31;
    const int hi   = lane >> 4;          // 0: lanes 0-15, 1: lanes 16-31
    const int lo   = lane & 15;
    const int wid  = (blockIdx.x * blockDim.x + threadIdx.x) >> 5; // [0,4096)

    const int pt  = wid & 255;           // pixel tile  [0,256)
    const int rem = wid >> 8;            // [0,16)
    const int mt  = rem & 3;             // M tile      [0,4)
    const int n   = rem >> 2;            // image       [0,4)

    const int p0 = pt << 4;
    const int m0 = mt << 4;

    const float* xn   = x + (size_t)n * 256 * HW;          // X[k][p] = xn[k*HW+p]
    const float* arow = Wc + (size_t)(m0 + lo) * 256;      // A: row M = m0+lo
    const float* bcol = xn + p0 + lo;                      // B: col N = p0+lo

    v8f acc = {};
    #pragma unroll 4
    for (int k0 = 0; k0 < 256; k0 += 4) {
        // A 16x4 f32: lane half 'hi' holds K = k0+2*hi, k0+2*hi+1 (contiguous b64)
        v2f a = *(const v2f*)(arow + k0 + 2 * hi);
        // B 4x16 f32: VGPR0 = row K=k0+hi, VGPR1 = row K=k0+2+hi (stride HW)
        v2f b;
        b.x = bcol[(size_t)(k0 + hi) * HW];
        b.y = bcol[(size_t)(k0 + 2 + hi) * HW];
        acc = __builtin_amdgcn_wmma_f32_16x16x4_f32(false, a, false, b,
                                                    (short)0, acc, false, false);
    }

    // D 16x16 f32: VGPR r -> M = m0 + r + 8*hi, N = p0 + lo
    float* cbase = comp + (size_t)n * 64 * HW + p0 + lo;
    #pragma unroll
    for (int r = 0; r < 8; ++r) {
        int m = m0 + r + 8 * hi;
        cbase[(size_t)m * HW] = acc[r] + bc[m];
    }
}

// ---------------------------------------------------------------------------
// [B] 3x3 conv encoder via implicit im2col, K tap-major: K index = t*64 + cc.
// Per tap: geometry/bounds/base computed once; 16 WMMA steps with two strided
// b32 B loads (constant-foldable offsets) + one contiguous b64 A load each.
// ---------------------------------------------------------------------------
__global__ __launch_bounds__(256) void encoder_wmma(
    const float* __restrict__ comp, const float* __restrict__ Wet,
    const float* __restrict__ be, float* __restrict__ m_out)
{
    const int lane = threadIdx.x & 31;
    const int hi   = lane >> 4;
    const int lo   = lane & 15;
    const int wid  = (blockIdx.x * blockDim.x + threadIdx.x) >> 5; // [0,7168)

    const int pt  = wid & 255;            // pixel tile [0,256)
    const int rem = wid >> 8;             // [0,28)
    const int mt  = rem % 7;              // M tile [0,7)
    const int n   = rem / 7;              // image

    const int p0 = pt << 4;
    const int e0 = mt << 4;

    const int p = p0 + lo;
    const int h = p >> 6;
    const int w = p & 63;

    const float* cn   = comp + (size_t)n * 64 * HW;
    const float* arow = Wet + (size_t)(e0 + lo) * 576;   // rows >=100 are zeros

    v8f acc = {};
    #pragma unroll
    for (int t = 0; t < 9; ++t) {
        const int dy = t / 3 - 1;          // constant after unroll
        const int dx = t % 3 - 1;
        const int hh = h + dy, ww = w + dx;
        const bool ok = ((unsigned)hh < 64u) && ((unsigned)ww < 64u);
        // clamped base keeps loads unconditional (EXEC stays all-ones for WMMA)
        const float* tb = cn + ((hh & 63) << 6) + (ww & 63);
        const float* at = arow + t * 64;

        #pragma unroll 4
        for (int cc0 = 0; cc0 < 64; cc0 += 4) {
            v2f a = *(const v2f*)(at + cc0 + 2 * hi);
            float b0 = tb[(size_t)(cc0 + hi) * HW];
            float b1 = tb[(size_t)(cc0 + 2 + hi) * HW];
            v2f b;
            b.x = ok ? b0 : 0.0f;
            b.y = ok ? b1 : 0.0f;
            acc = __builtin_amdgcn_wmma_f32_16x16x4_f32(false, a, false, b,
                                                        (short)0, acc, false, false);
        }
    }

    float* mbase = m_out + (size_t)n * 100 * HW + p0 + lo;
    #pragma unroll
    for (int r = 0; r < 8; ++r) {
        int e = e0 + r + 8 * hi;
        if (e < 100)
            mbase[(size_t)e * HW] = acc[r] + be[e];
    }
}

// ---------------------------------------------------------------------------
// [C] pixel-shuffle + softmax over the 25 kernel taps.
// m (N,100,64,64), channel = k*4 + i*2 + j  ->  mask (N,25,128,128).
// ---------------------------------------------------------------------------
__global__ __launch_bounds__(256) void softmax_mask(
    const float* __restrict__ m, float* __restrict__ mask)
{
    int idx = blockIdx.x * blockDim.x + threadIdx.x;    // [0, 4*128*128)
    int wo = idx & 127;
    int ho = (idx >> 7) & 127;
    int n  = idx >> 14;

    int i = ho & 1, j = wo & 1;
    int h = ho >> 1, w = wo >> 1;

    const float* mb = m + (size_t)n * 100 * HW + (size_t)(i * 2 + j) * HW
                        + (h << 6) + w;

    float v[25];
    float mx = -INFINITY;
    #pragma unroll
    for (int k = 0; k < 25; ++k) {
        v[k] = mb[(size_t)k * 4 * HW];
        mx = fmaxf(mx, v[k]);
    }
    float s = 0.0f;
    #pragma unroll
    for (int k = 0; k < 25; ++k) {
        v[k] = __expf(v[k] - mx);
        s += v[k];
    }
    float inv = 1.0f / s;

    float* ob = mask + (size_t)n * 25 * HOWO + (ho << 7) + wo;
    #pragma unroll
    for (int k = 0; k < 25; ++k)
        ob[(size_t)k * HOWO] = v[k] * inv;
}

// ---------------------------------------------------------------------------
// [D] reassembly: out[n,c,ho,wo] = sum_{ky,kx} mask[n,k,ho,wo] * x[n,c,h+ky-2,w+kx-2]
// One block per (n, ho) row; the 128x25 mask slab is staged in LDS; each lane
// keeps its 25 weights in VGPRs across the channel loop. Stores coalesce to
// 128B/wave (32 lanes x consecutive wo).
// ---------------------------------------------------------------------------
__global__ __launch_bounds__(256) void reassemble(
    const float* __restrict__ x, const float* __restrict__ mask,
    float* __restrict__ out)
{
    __shared__ float smask[128 * 25];

    const int ho = blockIdx.x & 127;
    const int n  = blockIdx.x >> 7;
    const int tid = threadIdx.x;

    const float* mb = mask + (size_t)n * 25 * HOWO + (ho << 7);
    for (int t = tid; t < 128 * 25; t += 256) {
        int k  = t >> 7;          // [0,25)
        int wo = t & 127;
        smask[wo * 25 + k] = mb[(size_t)k * HOWO + wo];
    }
    __syncthreads();

    const int wo    = tid & 127;
    const int cBase = tid >> 7;        // 0 or 1
    const int h = ho >> 1;
    const int w = wo >> 1;

    const float* xn = x + (size_t)n * 256 * HW;
    float* on = out + (size_t)n * 256 * HOWO + (ho << 7) + wo;

    float mk[25];
    #pragma unroll
    for (int k = 0; k < 25; ++k) mk[k] = smask[wo * 25 + k];

    for (int c = cBase; c < 256; c += 2) {
        const float* xc = xn + ((size_t)c << 12);
        float acc = 0.0f;
        #pragma unroll
        for (int ky = 0; ky < 5; ++ky) {
            int hh = h + ky - 2;
            bool rowok = (unsigned)hh < 64u;
            const float* xr = xc + ((hh & 63) << 6);
            #pragma unroll
            for (int kx = 0; kx < 5; ++kx) {
                int ww = w + kx - 2;
                bool ok = rowok && ((unsigned)ww < 64u);
                float xv = xr[ww & 63];
                acc = fmaf(ok ? xv : 0.0f, mk[ky * 5 + kx], acc);
            }
        }
        on[(size_t)c * HOWO] = acc;
    }
}

// ---------------------------------------------------------------------------
// Host-side orchestration. Workspace layout (bytes):
//   comp : [0,            4,194,304)   (4*64*64*64 f32)
//   m    : [4,194,304,   10,747,904)   (4*100*64*64 f32)
//   mask : [10,747,904,  17,301,504)   (4*25*128*128 f32)
//   Wet  : [17,301,504,  17,559,552)   (112*576 f32, zero-padded rows)
// ---------------------------------------------------------------------------
extern "C" void kernel_launch(void* const* d_in, const int* in_sizes, int n_in,
                              void* d_out, int out_size, void* d_ws, size_t ws_size,
                              hipStream_t stream)
{
    const float* x  = (const float*)d_in[0];   // (4,256,64,64)
    const float* Wc = (const float*)d_in[1];   // (64,256,1,1)
    const float* bc = (const float*)d_in[2];   // (64,)
    const float* We = (const float*)d_in[3];   // (100,64,3,3)
    const float* be = (const float*)d_in[4];   // (100,)
    float* out = (float*)d_out;                // (4,256,128,128)

    char* ws = (char*)d_ws;
    float* comp = (float*)(ws);
    float* mbuf = (float*)(ws + 4194304);
    float* mask = (float*)(ws + 10747904);
    float* Wet  = (float*)(ws + 17301504);

    // [T] 112*576 = 64512 elements
    transpose_we<<<252, 256, 0, stream>>>(We, Wet);
    // [A] 4096 waves = 131072 threads
    compress_wmma<<<512, 256, 0, stream>>>(x, Wc, bc, comp);
    // [B] 7168 waves = 229376 threads
    encoder_wmma<<<896, 256, 0, stream>>>(comp, Wet, be, mbuf);
    // [C] 65536 output pixels
    softmax_mask<<<256, 256, 0, stream>>>(mbuf, mask);
    // [D] one block per (n, ho)
    reassemble<<<512, 256, 0, stream>>>(x, mask, out);
}